// FixedVisionAttention_46385646797370
// MI455X (gfx1250) — compile-verified
//
#include <hip/hip_runtime.h>
#include <hip/hip_bf16.h>

// ---------------------------------------------------------------------------
// FixedVisionAttention for MI455X (gfx1250), all-WMMA fp32 pipeline.
//   qkv = hs @ qkv_w + b       (WMMA f32 16x16x4, async-LDS double buffering)
//   rope(q,k) in place
//   per (seg,head): flash attention with WMMA QK^T and PV
//   out = attn @ proj_w + b    (same GEMM)
// ---------------------------------------------------------------------------

typedef float v2f __attribute__((ext_vector_type(2)));
typedef float v8f __attribute__((ext_vector_type(8)));

#define S_TOT   8192
#define DIMM    1280
#define NHEAD   16
#define HDIM    80
#define NSEG    8
#define SEGL    1024
#define QKVN    3840
#define KSTEP   32

// Numeric value of an AS(3) pointer is the LDS byte offset.
__device__ __forceinline__ unsigned lds_byte_off(const void* p)
{
    return (unsigned)(unsigned long long)(__attribute__((address_space(3))) const void*)p;
}

// CDNA5 async copy: global -> LDS, 16B per lane, tracked by ASYNCcnt.
__device__ __forceinline__ void async_copy_b128(unsigned lds_off, const float* gptr)
{
    asm volatile("global_load_async_to_lds_b128 %0, %1, off"
                 :: "v"(lds_off), "v"(gptr) : "memory");
}

// ---------------------------------------------------------------------------
// GEMM: C[M,N] = A[M,K] @ B[K,N] + bias[N].
// Block = 4 waves (128 thr), tile 128x64; wave w owns rows [32w, 32w+32),
// all 4 col sub-tiles (two 16-row C groups share every B fragment).
// B k-panels (32 x 64 f32, 8KB) double-buffered in LDS via async-to-LDS.
// ---------------------------------------------------------------------------
__global__ __launch_bounds__(128)
void gemm_wmma_f32(const float* __restrict__ A, const float* __restrict__ B,
                   const float* __restrict__ bias, float* __restrict__ C,
                   int M, int N, int K)
{
    __shared__ float Bs[2][KSTEP * 64];
    const int t    = threadIdx.x;
    const int wave = t >> 5;
    const int lane = t & 31;
    const int lo   = lane & 15;   // N / M index within fragment
    const int hi   = lane >> 4;   // half-wave select
    const int row0 = blockIdx.y * 128 + wave * 32;
    const int col0 = blockIdx.x * 64;

    const unsigned lds0 = lds_byte_off(&Bs[0][0]);

    v8f acc[2][4] = {};
    const float* Ar0 = A + (size_t)(row0 + lo) * K;
    const float* Ar1 = A + (size_t)(row0 + 16 + lo) * K;

    // Issue one 32x64 B panel (2048 floats): 4 x b128 per thread, 16B aligned
    // in both global (row-major, 4 | col offset) and LDS.
    auto issue_panel = [&](int k0, int buf) {
        #pragma unroll
        for (int i = 0; i < 4; ++i) {
            int f  = i * 512 + t * 4;                  // 0..2047
            int kk = f >> 6, nn = f & 63;
            const float* g = B + (size_t)(k0 + kk) * N + col0 + nn;
            async_copy_b128(lds0 + (unsigned)(buf * (KSTEP * 64) + f) * 4u, g);
        }
    };

    issue_panel(0, 0);
    const int NK = K / KSTEP;
    for (int s = 0; s < NK; ++s) {
        const int cur = s & 1;
        if (s + 1 < NK) {
            issue_panel((s + 1) * KSTEP, cur ^ 1);     // prefetch next panel
            // async ops complete in order: <=4 outstanding => panel s landed
            asm volatile("s_wait_asynccnt 0x4" ::: "memory");
        } else {
            asm volatile("s_wait_asynccnt 0x0" ::: "memory");
        }
        __syncthreads();                               // panel s visible to all waves

        const float* Bp = &Bs[cur][0];
        #pragma unroll
        for (int c = 0; c < 8; ++c) {
            int kc = s * KSTEP + 4 * c + 2 * hi;
            // A fragments: lane holds A[row][kc..kc+1] (contiguous b64 load)
            v2f a0 = *(const v2f*)(Ar0 + kc);
            v2f a1 = *(const v2f*)(Ar1 + kc);
            #pragma unroll
            for (int j = 0; j < 4; ++j) {
                v2f b;
                b.x = Bp[(4 * c + 2 * hi)     * 64 + 16 * j + lo];
                b.y = Bp[(4 * c + 2 * hi + 1) * 64 + 16 * j + lo];
                acc[0][j] = __builtin_amdgcn_wmma_f32_16x16x4_f32(
                    false, a0, false, b, (short)0, acc[0][j], false, false);
                acc[1][j] = __builtin_amdgcn_wmma_f32_16x16x4_f32(
                    false, a1, false, b, (short)0, acc[1][j], false, false);
            }
        }
        __syncthreads();   // all waves done reading before buffer is re-filled
    }

    #pragma unroll
    for (int j = 0; j < 4; ++j) {
        float bv = bias[col0 + 16 * j + lo];
        #pragma unroll
        for (int v = 0; v < 8; ++v) {
            int r = row0 + v + 8 * hi;     // C layout: M = v + 8*hi, N = lo
            C[(size_t)r        * N + col0 + 16 * j + lo] = acc[0][j][v] + bv;
            C[(size_t)(r + 16) * N + col0 + 16 * j + lo] = acc[1][j][v] + bv;
        }
    }
}

// ---------------------------------------------------------------------------
// RoPE in place on q and k planes of qkv.  One thread per (token, head,
// rotation pair) so the (d, d+40) pair is read+written by a single thread.
// ---------------------------------------------------------------------------
__global__ __launch_bounds__(256)
void rope_kernel(float* __restrict__ qkv, const float* __restrict__ rope)
{
    int idx = blockIdx.x * blockDim.x + threadIdx.x;   // S * 16 * 40
    if (idx >= S_TOT * NHEAD * 40) return;
    int s  = idx / (NHEAD * 40);
    int r  = idx - s * (NHEAD * 40);
    int h  = r / 40;
    int dp = r - h * 40;

    float fr = rope[s * 40 + dp];
    float cs = cosf(fr), sn = sinf(fr);

    size_t bq = (size_t)s * QKVN + h * HDIM;   // q plane
    #pragma unroll
    for (int plane = 0; plane < 2; ++plane) {  // q then k
        size_t base = bq + plane * DIMM;
        float x1 = qkv[base + dp];
        float x2 = qkv[base + dp + 40];
        qkv[base + dp]      = x1 * cs - x2 * sn;   // x*cos + rot_half(x)*sin
        qkv[base + dp + 40] = x2 * cs + x1 * sn;
    }
}

// ---------------------------------------------------------------------------
// Flash attention: one wave (32 threads) per (seg, head, 16-row q block).
// Q fragments (pre-scaled by 1/sqrt(80)) live in registers; K^T fragments
// are contiguous float2 loads from row-major K; P is transposed through LDS
// (C-layout -> A-layout), ordering guaranteed by in-order DS + s_wait_dscnt.
// ---------------------------------------------------------------------------
__global__ __launch_bounds__(32)
void attn_kernel(const float* __restrict__ qkv, float* __restrict__ out)
{
    __shared__ float Ps[16 * 16];
    const int lane = threadIdx.x & 31;
    const int lo   = lane & 15;
    const int hi   = lane >> 4;
    const int qb   = blockIdx.x;        // 0..63
    const int h    = blockIdx.y;        // 0..15
    const int seg  = blockIdx.z;        // 0..7
    const int tok0 = seg * SEGL + qb * 16;

    const float scale = 0.11180339887498949f;  // 1/sqrt(80)
    const float* Qb = qkv + (size_t)tok0 * QKVN + h * HDIM;
    const float* Kb = qkv + (size_t)(seg * SEGL) * QKVN + DIMM + h * HDIM;
    const float* Vb = Kb + DIMM;

    // Q A-fragments: 20 chunks of K=4
    v2f aq[20];
    #pragma unroll
    for (int c = 0; c < 20; ++c) {
        v2f a = *(const v2f*)(Qb + (size_t)lo * QKVN + 4 * c + 2 * hi);
        aq[c].x = a.x * scale;
        aq[c].y = a.y * scale;
    }

    v8f   O[5] = {};
    float rmax[8], rsum[8];
    #pragma unroll
    for (int v = 0; v < 8; ++v) { rmax[v] = -1e30f; rsum[v] = 0.0f; }

    for (int j = 0; j < 64; ++j) {
        // ---- S tile = (Q*scale) @ K_j^T ------------------------------------
        v8f s = {};
        const float* Kj = Kb + (size_t)(j * 16 + lo) * QKVN + 2 * hi;
        #pragma unroll
        for (int c = 0; c < 20; ++c) {
            v2f b = *(const v2f*)(Kj + 4 * c);     // K^T frag: contiguous
            s = __builtin_amdgcn_wmma_f32_16x16x4_f32(
                false, aq[c], false, b, (short)0, s, false, false);
        }

        // ---- online softmax (row reductions via 16-lane butterflies) -------
        float corr[8];
        #pragma unroll
        for (int v = 0; v < 8; ++v) {
            float m = s[v];
            m = fmaxf(m, __shfl_xor(m, 1, 32));
            m = fmaxf(m, __shfl_xor(m, 2, 32));
            m = fmaxf(m, __shfl_xor(m, 4, 32));
            m = fmaxf(m, __shfl_xor(m, 8, 32));
            float nm = fmaxf(rmax[v], m);
            float p  = __expf(s[v] - nm);
            float ps = p;
            ps += __shfl_xor(ps, 1, 32);
            ps += __shfl_xor(ps, 2, 32);
            ps += __shfl_xor(ps, 4, 32);
            ps += __shfl_xor(ps, 8, 32);
            corr[v] = __expf(rmax[v] - nm);
            rsum[v] = rsum[v] * corr[v] + ps;
            rmax[v] = nm;
            Ps[(v + 8 * hi) * 16 + lo] = p;        // P in row-major LDS
        }
        #pragma unroll
        for (int g = 0; g < 5; ++g)
            #pragma unroll
            for (int v = 0; v < 8; ++v) O[g][v] *= corr[v];

        // DS is in-order within a wave; make the store->load turnaround explicit
        asm volatile("s_wait_dscnt 0" ::: "memory");

        // ---- P: C-layout -> A-layout through LDS ---------------------------
        v2f ap[4];
        #pragma unroll
        for (int c = 0; c < 4; ++c) {
            ap[c].x = Ps[lo * 16 + 4 * c + 2 * hi];
            ap[c].y = Ps[lo * 16 + 4 * c + 2 * hi + 1];
        }

        // ---- O += P @ V_j --------------------------------------------------
        const float* Vj = Vb + (size_t)(j * 16) * QKVN;
        #pragma unroll
        for (int g = 0; g < 5; ++g) {
            #pragma unroll
            for (int c = 0; c < 4; ++c) {
                v2f b;
                b.x = Vj[(size_t)(4 * c + 2 * hi)     * QKVN + g * 16 + lo];
                b.y = Vj[(size_t)(4 * c + 2 * hi + 1) * QKVN + g * 16 + lo];
                O[g] = __builtin_amdgcn_wmma_f32_16x16x4_f32(
                    false, ap[c], false, b, (short)0, O[g], false, false);
            }
        }
    }

    // ---- normalize and write (S, H*D) row-major ----------------------------
    #pragma unroll
    for (int g = 0; g < 5; ++g)
        #pragma unroll
        for (int v = 0; v < 8; ++v) {
            int tok = tok0 + v + 8 * hi;
            out[(size_t)tok * DIMM + h * HDIM + g * 16 + lo] = O[g][v] / rsum[v];
        }
}

// ---------------------------------------------------------------------------
extern "C" void kernel_launch(void* const* d_in, const int* in_sizes, int n_in,
                              void* d_out, int out_size, void* d_ws, size_t ws_size,
                              hipStream_t stream)
{
    const float* hidden = (const float*)d_in[0];
    const float* rope   = (const float*)d_in[1];
    const float* qkv_w  = (const float*)d_in[2];
    const float* qkv_b  = (const float*)d_in[3];
    const float* proj_w = (const float*)d_in[4];
    const float* proj_b = (const float*)d_in[5];
    // d_in[6] = cu_seqlens: uniform 1024-token segments per the reference constants.

    float* qkv  = (float*)d_ws;                          // 8192 x 3840 f32 (126 MB)
    float* attn = qkv + (size_t)S_TOT * QKVN;            // 8192 x 1280 f32 ( 42 MB)
    float* outp = (float*)d_out;

    // 1) QKV projection
    gemm_wmma_f32<<<dim3(QKVN / 64, S_TOT / 128), dim3(128), 0, stream>>>(
        hidden, qkv_w, qkv_b, qkv, S_TOT, QKVN, DIMM);

    // 2) RoPE on q,k planes (in place)
    int rope_threads = S_TOT * NHEAD * 40;
    rope_kernel<<<(rope_threads + 255) / 256, 256, 0, stream>>>(qkv, rope);

    // 3) Segment-local attention
    attn_kernel<<<dim3(SEGL / 16, NHEAD, NSEG), dim3(32), 0, stream>>>(qkv, attn);

    // 4) Output projection
    gemm_wmma_f32<<<dim3(DIMM / 64, S_TOT / 128), dim3(128), 0, stream>>>(
        attn, proj_w, proj_b, outp, S_TOT, DIMM, DIMM);
}